// Recommender_72799695667431
// MI455X (gfx1250) — compile-verified
//
#include <hip/hip_runtime.h>

#define N_ENT   200000
#define N_USERS 100000
#define N_REL   20
#define EMB     64
#define N_EDGES 1200000
#define NNZV    1000000

typedef float v2f __attribute__((ext_vector_type(2)));
typedef float v8f __attribute__((ext_vector_type(8)));

__device__ __forceinline__ float lane_read(float v, int srcLane) {
  return __int_as_float(__builtin_amdgcn_ds_bpermute(srcLane << 2, __float_as_int(v)));
}

// ---------------------------------------------------------------------------
// Intent refinement: i_new[h] = 0.5*(mean_j softmax_j(<intent_h, r_j>) * r_j + intent_h)
// One wave; lane owns dims (lane, lane+32). Butterfly reductions via ds_bpermute.
// ---------------------------------------------------------------------------
__global__ __launch_bounds__(32) void intent_kernel(const float* __restrict__ intent_emb,
                                                    const float* __restrict__ r_emb,
                                                    float* __restrict__ i_new) {
  int lane = threadIdx.x;
  for (int h = 0; h < 5; ++h) {
    int base = (h == 0) ? 0 : (h - 1) * 5;
    int nr   = (h == 0) ? 20 : 5;
    float iv0 = intent_emb[h * EMB + lane];
    float iv1 = intent_emb[h * EMB + lane + 32];
    float dots[20];
    for (int j = 0; j < nr; ++j) {
      int rj = base + j;
      float p = iv0 * r_emb[rj * EMB + lane] + iv1 * r_emb[rj * EMB + lane + 32];
      for (int off = 16; off > 0; off >>= 1) p += lane_read(p, lane ^ off);
      dots[j] = p;
    }
    float mx = -1e30f;
    for (int j = 0; j < nr; ++j) mx = fmaxf(mx, dots[j]);
    float den = 0.f;
    for (int j = 0; j < nr; ++j) { dots[j] = __expf(dots[j] - mx); den += dots[j]; }
    float o0 = 0.f, o1 = 0.f;
    for (int j = 0; j < nr; ++j) {
      int rj = base + j;
      float a = dots[j] / den;
      o0 += a * r_emb[rj * EMB + lane];
      o1 += a * r_emb[rj * EMB + lane + 32];
    }
    float invn = 1.0f / (float)nr;
    i_new[h * EMB + lane]      = (o0 * invn + iv0) * 0.5f;
    i_new[h * EMB + lane + 32] = (o1 * invn + iv1) * 0.5f;
  }
}

// ---------------------------------------------------------------------------
// In-degree counts (hop-invariant).
// ---------------------------------------------------------------------------
__global__ void count_kernel(const int* __restrict__ head, float* __restrict__ cnt, int nE) {
  int e = blockIdx.x * blockDim.x + threadIdx.x;
  if (e < nE) atomicAdd(cnt + head[e], 1.0f);
}

// ---------------------------------------------------------------------------
// KG edge scatter: s[head] += e[tail] * r[type-1]. 16 lanes per edge, float4 each.
// Accumulator (51 MB) is L2-resident -> atomics stay on-chip.
// ---------------------------------------------------------------------------
__global__ void edge_scatter(const float4* __restrict__ e4, const float4* __restrict__ r4,
                             const int* __restrict__ head, const int* __restrict__ tail,
                             const int* __restrict__ type, float* __restrict__ s, int nE) {
  unsigned gid = blockIdx.x * blockDim.x + threadIdx.x;
  unsigned e = gid >> 4;
  if (e >= (unsigned)nE) return;
  int li = gid & 15;
  int h = head[e], t = tail[e], ty = type[e] - 1;
  float4 ev = e4[(size_t)t * 16 + li];
  float4 rv = r4[(size_t)ty * 16 + li];
  float* dst = s + (size_t)h * EMB + li * 4;
  atomicAdd(dst + 0, ev.x * rv.x);
  atomicAdd(dst + 1, ev.y * rv.y);
  atomicAdd(dst + 2, ev.z * rv.z);
  atomicAdd(dst + 3, ev.w * rv.w);
}

// ---------------------------------------------------------------------------
// User interaction scatter: uacc[row] += val * e[col].
// ---------------------------------------------------------------------------
__global__ void interact_scatter(const float4* __restrict__ e4, const float* __restrict__ vals,
                                 const int* __restrict__ rows, const int* __restrict__ cols,
                                 float* __restrict__ uacc, int nnz) {
  unsigned gid = blockIdx.x * blockDim.x + threadIdx.x;
  unsigned i = gid >> 4;
  if (i >= (unsigned)nnz) return;
  int li = gid & 15;
  float v = vals[i];
  int r = rows[i], c = cols[i];
  float4 ev = e4[(size_t)c * 16 + li];
  float* dst = uacc + (size_t)r * EMB + li * 4;
  atomicAdd(dst + 0, v * ev.x);
  atomicAdd(dst + 1, v * ev.y);
  atomicAdd(dst + 2, v * ev.z);
  atomicAdd(dst + 3, v * ev.w);
}

// ---------------------------------------------------------------------------
// entity_agg = s / max(cnt,1)
// ---------------------------------------------------------------------------
__global__ void entity_finalize(const float4* __restrict__ s4, const float* __restrict__ cnt,
                                float4* __restrict__ out4, int nent) {
  unsigned gid = blockIdx.x * blockDim.x + threadIdx.x;
  if (gid >= (unsigned)(nent * 16)) return;
  float inv = 1.0f / fmaxf(cnt[gid >> 4], 1.0f);
  float4 v = s4[gid];
  v.x *= inv; v.y *= inv; v.z *= inv; v.w *= inv;
  out4[gid] = v;
}

// ---------------------------------------------------------------------------
// Per 16-user tile (one wave):
//   GEMM1 (WMMA f32 16x16x4): C = U_tile(16x64) @ i_new^T(64x16) [5 valid cols]
//   cross-lane softmax over 5 intents (ds_bpermute)
//   GEMM2 (WMMA f32 16x16x4): mult = score(16x8pad) @ i_new(8x64)
//   u_next = uacc * (mult + 1)
// f32 A frag:  row = lane&15, K = kb + 2*(lane>>4) + vgpr
// f32 B/C/D:   col = lane&15, row = vgpr + half*(lane>>4)
// ---------------------------------------------------------------------------
__global__ __launch_bounds__(32) void user_wmma_kernel(const float* __restrict__ u_cur,
                                                       const float* __restrict__ i_new,
                                                       const float* __restrict__ uacc,
                                                       float* __restrict__ u_next) {
  __shared__ float sI[5 * EMB];   // i_new, row-major [5][64]
  __shared__ float sS[16 * 8];    // normalized scores, padded K=8

  int lane = threadIdx.x;
  int g = lane >> 4;     // lane half
  int n = lane & 15;     // B/C/D column; also A row
  int row0 = blockIdx.x * 16;

  for (int i = lane; i < 5 * EMB; i += 32) sI[i] = i_new[i];
  asm volatile("s_wait_dscnt 0" ::: "memory");

  // ---- GEMM1: C[m][n] = sum_k U[row0+m][k] * Inew[n][k]
  v8f c = {};
  const float* urow = u_cur + (size_t)(row0 + n) * EMB;
  int nn = (n < 5) ? n : 0;
  float bmask = (n < 5) ? 1.0f : 0.0f;
  for (int kb = 0; kb < EMB; kb += 4) {
    int ka = kb + 2 * g;
    v2f a, b;
    a.x = urow[ka];
    a.y = urow[ka + 1];
    b.x = bmask * sI[nn * EMB + ka];
    b.y = bmask * sI[nn * EMB + ka + 1];
    c = __builtin_amdgcn_wmma_f32_16x16x4_f32(false, a, false, b, (short)0, c, false, false);
  }

  // ---- softmax over the 5 intents of each row (values live in lanes g*16+0..4)
#pragma unroll
  for (int v = 0; v < 8; ++v) {
    float val = c[v];
    int b0 = g * 16;
    float s0 = lane_read(val, b0 + 0);
    float s1 = lane_read(val, b0 + 1);
    float s2 = lane_read(val, b0 + 2);
    float s3 = lane_read(val, b0 + 3);
    float s4 = lane_read(val, b0 + 4);
    float mx = fmaxf(fmaxf(fmaxf(s0, s1), fmaxf(s2, s3)), s4);
    float den = __expf(s0 - mx) + __expf(s1 - mx) + __expf(s2 - mx) +
                __expf(s3 - mx) + __expf(s4 - mx);
    float e = (n < 5) ? (__expf(val - mx) / den) : 0.0f;
    if (n < 8) sS[(v + 8 * g) * 8 + n] = e;   // n in [5,8): zero padding for K=8
  }
  asm volatile("s_wait_dscnt 0" ::: "memory");

  // ---- GEMM2: mult = score(16x8) @ Inew(8x64), fused output
#pragma unroll
  for (int nt = 0; nt < 4; ++nt) {
    v8f d = {};
#pragma unroll
    for (int t = 0; t < 2; ++t) {
      int ka = t * 4 + 2 * g;
      int col = nt * 16 + n;
      v2f a, b;
      a.x = sS[n * 8 + ka];
      a.y = sS[n * 8 + ka + 1];
      int k0 = (ka < 5) ? ka : 0;
      int k1 = (ka + 1 < 5) ? (ka + 1) : 0;
      b.x = ((ka < 5) ? 1.0f : 0.0f) * sI[k0 * EMB + col];
      b.y = ((ka + 1 < 5) ? 1.0f : 0.0f) * sI[k1 * EMB + col];
      d = __builtin_amdgcn_wmma_f32_16x16x4_f32(false, a, false, b, (short)0, d, false, false);
    }
#pragma unroll
    for (int v = 0; v < 8; ++v) {
      size_t idx = (size_t)(row0 + v + 8 * g) * EMB + (nt * 16 + n);
      u_next[idx] = uacc[idx] * (d[v] + 1.0f);
    }
  }
}

// ---------------------------------------------------------------------------
extern "C" void kernel_launch(void* const* d_in, const int* in_sizes, int n_in,
                              void* d_out, int out_size, void* d_ws, size_t ws_size,
                              hipStream_t stream) {
  const float* entity_emb = (const float*)d_in[0];
  const float* user_emb   = (const float*)d_in[1];
  const float* intent_emb = (const float*)d_in[2];
  const float* r_emb      = (const float*)d_in[3];
  const float* ivals      = (const float*)d_in[4];
  const int*   e_head     = (const int*)d_in[5];
  const int*   e_tail     = (const int*)d_in[6];
  const int*   e_type     = (const int*)d_in[7];
  const int*   i_rows     = (const int*)d_in[8];
  const int*   i_cols     = (const int*)d_in[9];

  float* e_out = (float*)d_out;                          // [N_ENT, 64]
  float* u_out = (float*)d_out + (size_t)N_ENT * EMB;    // [N_USERS, 64]

  const size_t S_BYTES = (size_t)N_ENT * EMB * sizeof(float);    // 51.2 MB
  const size_t C_BYTES = (size_t)N_ENT * sizeof(float);          // 0.8 MB
  const size_t U_BYTES = (size_t)N_USERS * EMB * sizeof(float);  // 25.6 MB

  char* ws = (char*)d_ws;
  float* i_new = (float*)ws;                                   // 1.25 KB (pad 4 KB)
  float* sbuf  = (float*)(ws + 4096);
  float* cntf  = (float*)(ws + 4096 + S_BYTES);
  float* uacc  = (float*)(ws + 4096 + S_BYTES + C_BYTES);
  float* u1    = (float*)(ws + 4096 + S_BYTES + C_BYTES + U_BYTES);

  // hop-invariant pieces
  intent_kernel<<<1, 32, 0, stream>>>(intent_emb, r_emb, i_new);
  hipMemsetAsync(cntf, 0, C_BYTES, stream);
  count_kernel<<<(N_EDGES + 255) / 256, 256, 0, stream>>>(e_head, cntf, N_EDGES);

  const float* e_cur = entity_emb;
  const float* u_cur = user_emb;
  for (int hop = 0; hop < 2; ++hop) {
    hipMemsetAsync(sbuf, 0, S_BYTES, stream);
    hipMemsetAsync(uacc, 0, U_BYTES, stream);

    edge_scatter<<<(N_EDGES * 16 + 255) / 256, 256, 0, stream>>>(
        (const float4*)e_cur, (const float4*)r_emb, e_head, e_tail, e_type, sbuf, N_EDGES);
    interact_scatter<<<(NNZV * 16 + 255) / 256, 256, 0, stream>>>(
        (const float4*)e_cur, ivals, i_rows, i_cols, uacc, NNZV);

    entity_finalize<<<(N_ENT * 16 + 255) / 256, 256, 0, stream>>>(
        (const float4*)sbuf, cntf, (float4*)e_out, N_ENT);

    float* u_next = (hop == 0) ? u1 : u_out;
    user_wmma_kernel<<<N_USERS / 16, 32, 0, stream>>>(u_cur, i_new, uacc, u_next);

    e_cur = e_out;   // entity ping-pongs through d_out (cross-kernel, stream-ordered)
    u_cur = u_next;  // user goes through ws then d_out (never intra-kernel aliased)
  }
}